// QuaternionAttentionLNS_944892805665
// MI455X (gfx1250) — compile-verified
//
#include <hip/hip_runtime.h>

// Problem constants (from reference): B=4, S=128, D=256, quat=4, N=D*4=1024
#define BB 4
#define SS 128
#define DD 256
#define NN 1024   // D*4, flattened feature dim
#define EPS 1e-6f

typedef __attribute__((ext_vector_type(2))) float v2f;
typedef __attribute__((ext_vector_type(8))) float v8f;

// ---------------------------------------------------------------------------
// Kernel 1: real-space LNS linear.
// out[b,s,n] = max(x[b,s,c],eps) + exp(w[c,s,d]) + exp(bias[d,c]),  n = d*4+c
// (exp(logaddexp(...)) == sum of exps; all terms >= eps so later clips vanish)
// ---------------------------------------------------------------------------
__global__ void lns_linear_kernel(const float* __restrict__ x,
                                  const float* __restrict__ w,
                                  const float* __restrict__ bias,
                                  float* __restrict__ out) {
    int idx = blockIdx.x * blockDim.x + threadIdx.x;   // 0 .. B*S*N-1
    int n  = idx & (NN - 1);
    int bs = idx >> 10;                                // b*S + s
    int s  = bs & (SS - 1);
    int b  = bs >> 7;
    int d  = n >> 2;
    int c  = n & 3;
    float val = fmaxf(x[(b * SS + s) * 4 + c], EPS)
              + __expf(w[(c * SS + s) * DD + d])
              + __expf(bias[d * 4 + c]);
    out[idx] = val;
}

// ---------------------------------------------------------------------------
// Kernel 2: scores[b,i,j] = sum_n Qr[b,i,n] * Kr[b,j,n]
// One wave32 per 16x16 output tile; K-loop over N=1024 in steps of 4 using
// V_WMMA_F32_16X16X4_F32.
// A fragment (16x4): lanes 0-15 hold row M=lane, K={0,1}; lanes 16-31 K={2,3}.
// B fragment (4x16): lanes 0-15 hold col N=lane, K={0,1}; lanes 16-31 K={2,3}.
// B[k][j] = Kr[j][k] -> per-lane contiguous float2 from Kr row j.
// ---------------------------------------------------------------------------
__global__ void scores_wmma_kernel(const float* __restrict__ Qr,
                                   const float* __restrict__ Kr,
                                   float* __restrict__ scores) {
    int wave = (blockIdx.x * blockDim.x + threadIdx.x) >> 5;  // 0..255
    int lane = threadIdx.x & 31;
    int b  = wave >> 6;
    int i0 = ((wave >> 3) & 7) << 4;
    int j0 = (wave & 7) << 4;
    int half = lane >> 4;      // 0: K 0-1, 1: K 2-3
    int lr   = lane & 15;

    const float* arow = Qr + (size_t)(b * SS + i0 + lr) * NN + 2 * half;
    const float* brow = Kr + (size_t)(b * SS + j0 + lr) * NN + 2 * half;

    v8f c = {};
    #pragma unroll 4
    for (int k = 0; k < NN; k += 4) {
        v2f a  = *(const v2f*)(arow + k);
        v2f bm = *(const v2f*)(brow + k);
        c = __builtin_amdgcn_wmma_f32_16x16x4_f32(false, a, false, bm,
                                                  (short)0, c, false, false);
    }

    // C/D layout: VGPR r holds M = r + 8*half, N = lr
    float* srow = scores + (size_t)(b * SS + i0 + 8 * half) * SS + j0 + lr;
    #pragma unroll
    for (int r = 0; r < 8; ++r) srow[(size_t)r * SS] = c[r];
}

// ---------------------------------------------------------------------------
// Kernel 3: in-place row softmax over j, then attn = max(softmax, eps).
// One block (128 threads) per (b,i) row.
// ---------------------------------------------------------------------------
__global__ void softmax_clip_kernel(float* __restrict__ scores) {
    __shared__ float red[SS];
    int row = blockIdx.x;         // b*S + i
    int t   = threadIdx.x;
    float v = scores[(size_t)row * SS + t];
    red[t] = v;
    __syncthreads();
    for (int off = SS / 2; off > 0; off >>= 1) {
        if (t < off) red[t] = fmaxf(red[t], red[t + off]);
        __syncthreads();
    }
    float m = red[0];
    __syncthreads();
    float e = __expf(v - m);
    red[t] = e;
    __syncthreads();
    for (int off = SS / 2; off > 0; off >>= 1) {
        if (t < off) red[t] += red[t + off];
        __syncthreads();
    }
    float sum = red[0];
    scores[(size_t)row * SS + t] = fmaxf(e / sum, EPS);
}

// ---------------------------------------------------------------------------
// Kernel 4: out[b,i,n] = sum_j attn[b,i,j] * Vr[b,j,n]
// One wave per 16(i) x 16(n) tile; K-loop over j (128) in steps of 4.
// A = attn (row-major over j -> contiguous float2 per lane).
// B[k=j][n] = Vr[j][n]: lane = n (coalesced), two strided loads for K pair.
// ---------------------------------------------------------------------------
__global__ void out_wmma_kernel(const float* __restrict__ attn,
                                const float* __restrict__ Vr,
                                float* __restrict__ out) {
    int wave = (blockIdx.x * blockDim.x + threadIdx.x) >> 5;  // 0..2047
    int lane = threadIdx.x & 31;
    int b  = wave >> 9;
    int i0 = ((wave >> 6) & 7) << 4;
    int n0 = (wave & 63) << 4;
    int half = lane >> 4;
    int lr   = lane & 15;

    const float* arow = attn + (size_t)(b * SS + i0 + lr) * SS + 2 * half;
    const float* bcol = Vr + (size_t)(b * SS) * NN + n0 + lr;

    v8f c = {};
    #pragma unroll 4
    for (int k = 0; k < SS; k += 4) {
        v2f a = *(const v2f*)(arow + k);
        int kk = k + 2 * half;
        v2f bm;
        bm[0] = bcol[(size_t)kk * NN];
        bm[1] = bcol[(size_t)(kk + 1) * NN];
        c = __builtin_amdgcn_wmma_f32_16x16x4_f32(false, a, false, bm,
                                                  (short)0, c, false, false);
    }

    float* orow = out + (size_t)(b * SS + i0 + 8 * half) * NN + n0 + lr;
    #pragma unroll
    for (int r = 0; r < 8; ++r) orow[(size_t)r * NN] = c[r];
}

// ---------------------------------------------------------------------------
extern "C" void kernel_launch(void* const* d_in, const int* in_sizes, int n_in,
                              void* d_out, int out_size, void* d_ws, size_t ws_size,
                              hipStream_t stream) {
    (void)in_sizes; (void)n_in; (void)out_size; (void)ws_size;

    const float* query  = (const float*)d_in[0];
    const float* key_in = (const float*)d_in[1];
    const float* value  = (const float*)d_in[2];
    const float* qw = (const float*)d_in[3];
    const float* qb = (const float*)d_in[4];
    const float* kw = (const float*)d_in[5];
    const float* kb = (const float*)d_in[6];
    const float* vw = (const float*)d_in[7];
    const float* vb = (const float*)d_in[8];
    float* out = (float*)d_out;

    // Workspace layout (floats): Qr | Kr | Vr | scores(->attn in place)
    float* Qr = (float*)d_ws;
    float* Kr = Qr + (size_t)BB * SS * NN;
    float* Vr = Kr + (size_t)BB * SS * NN;
    float* sc = Vr + (size_t)BB * SS * NN;   // [B,S,S]

    const int total = BB * SS * NN;          // 524288
    lns_linear_kernel<<<total / 256, 256, 0, stream>>>(query, qw, qb, Qr);
    lns_linear_kernel<<<total / 256, 256, 0, stream>>>(key_in, kw, kb, Kr);
    lns_linear_kernel<<<total / 256, 256, 0, stream>>>(value, vw, vb, Vr);

    // 256 tiles, 4 waves (128 thr) per block -> 64 blocks
    scores_wmma_kernel<<<64, 128, 0, stream>>>(Qr, Kr, sc);

    softmax_clip_kernel<<<BB * SS, SS, 0, stream>>>(sc);

    // 2048 tiles, 4 waves per block -> 512 blocks
    out_wmma_kernel<<<512, 128, 0, stream>>>(sc, Vr, out);
}